// RBFNetwork_17377437680027
// MI455X (gfx1250) — compile-verified
//
#include <hip/hip_runtime.h>

// CDNA5 / gfx1250 RBF network:
//   dist(n,c) = |x_n|^2 + |c|^2 - 2 x_n.c     -> via V_WMMA_F32_16X16X4_F32
//   rbf = exp2(-(beta*log2e)*dist)            -> v_exp_f32
//   out = (rbf @ w) / rowsum(rbf)
//
// A-row (16x4, K padded):  [x0, x1, x2, 1]
// B-col (4x16):            [-2c0, -2c1, -2c2, |c|^2]
// => D = -2 x.c + |c|^2 ; add |x|^2 per lane afterwards.

typedef float v2f __attribute__((ext_vector_type(2)));
typedef float v8f __attribute__((ext_vector_type(8)));

#define CMAX 2048                      // max centers staged in LDS (C=2000)
#define LOG2E 1.4426950408889634f

__global__ __launch_bounds__(256) void rbf_wmma_kernel(
    const float* __restrict__ x,        // (N,3)
    const float* __restrict__ centers,  // (C,3)
    const float* __restrict__ beta,     // (C,)
    const float* __restrict__ lin_w,    // (1,C) -> w[c]
    float* __restrict__ out,            // (N,1)
    int N, int C)
{
    // LDS-staged, pre-swizzled center data (48 KB total):
    __shared__ float2 s_lo[CMAX];   // (-2c0, -2c1)    -> B K=0,1 (lanes 0..15)
    __shared__ float2 s_hi[CMAX];   // (-2c2, |c|^2)   -> B K=2,3 (lanes 16..31)
    __shared__ float2 s_bw[CMAX];   // (beta*log2e, w)

    const int tid   = threadIdx.x;
    const int nblk  = (C + 15) >> 4;
    const int cpad  = nblk << 4;

    // ---- cooperative center prep into LDS ----
    for (int c = tid; c < cpad; c += blockDim.x) {
        if (c < C) {
            float c0 = centers[3*c+0];
            float c1 = centers[3*c+1];
            float c2 = centers[3*c+2];
            s_lo[c] = make_float2(-2.0f*c0, -2.0f*c1);
            s_hi[c] = make_float2(-2.0f*c2, c0*c0 + c1*c1 + c2*c2);
            s_bw[c] = make_float2(beta[c] * LOG2E, lin_w[c]);
        } else {
            // padding centers: dist -> huge, w -> 0  => rbf contributes 0
            s_lo[c] = make_float2(0.0f, 0.0f);
            s_hi[c] = make_float2(0.0f, 1.0e30f);
            s_bw[c] = make_float2(1.0f, 0.0f);
        }
    }
    __syncthreads();

    const int lane = tid & 31;
    const int wave = tid >> 5;
    const int half = lane >> 4;        // 0 => K=0,1 / rows 0..7 ; 1 => K=2,3 / rows 8..15
    const int col  = lane & 15;
    const int tile = blockIdx.x * 128 + wave * 16;   // 8 waves * 16 rows per block

    // ---- A fragment: row m = tile+col, elements [x0,x1,x2,1] ----
    int m  = tile + col;
    int mc = (m < N) ? m : 0;          // clamp (keeps EXEC all-ones for WMMA)
    float xa = x[3*mc+0];
    float xb = x[3*mc+1];
    float xcv = x[3*mc+2];
    v2f A;
    A.x = half ? xcv  : xa;            // lanes 0..15: K=0 ; lanes 16..31: K=2
    A.y = half ? 1.0f : xb;            // lanes 0..15: K=1 ; lanes 16..31: K=3 (pad=1)
    float x2own = xa*xa + xb*xb + xcv*xcv;

    // |x|^2 for the 8 rows covered by this lane's D fragment
    float x2v[8];
    #pragma unroll
    for (int i = 0; i < 8; ++i)
        x2v[i] = __shfl(x2own, half ? (8 + i) : i, 16);

    float ssum[8], sacc[8];
    #pragma unroll
    for (int i = 0; i < 8; ++i) { ssum[i] = 0.0f; sacc[i] = 0.0f; }

    const float2* __restrict__ bsel = half ? s_hi : s_lo;

    // ---- main loop: 16 centers per iteration, one WMMA each ----
    for (int j = 0; j < nblk; ++j) {
        const int cidx = (j << 4) + col;
        float2 bfrag = bsel[cidx];     // ds_load_b64, conflict-free
        float2 bw    = s_bw[cidx];     // (beta*log2e, w) broadcast per column

        v2f B;
        B.x = bfrag.x;                 // K=0 (lo) / K=2 (hi)
        B.y = bfrag.y;                 // K=1 (lo) / K=3 (hi)

        v8f cz = {};
        // (neg_a, A, neg_b, B, c_mod, C, reuse_a, reuse_b)
        v8f d = __builtin_amdgcn_wmma_f32_16x16x4_f32(
            false, A, false, B, (short)0, cz, false, false);

        #pragma unroll
        for (int i = 0; i < 8; ++i) {
            float dist = d[i] + x2v[i];                 // + |x|^2
            float t    = bw.x * dist;                   // beta*log2e*dist
            float e    = __builtin_amdgcn_exp2f(-t);    // v_exp_f32
            ssum[i] += e;
            sacc[i]  = fmaf(e, bw.y, sacc[i]);
        }
    }

    // ---- reduce across the 16 columns (butterfly within each half-wave) ----
    #pragma unroll
    for (int i = 0; i < 8; ++i) {
        #pragma unroll
        for (int mask = 1; mask < 16; mask <<= 1) {
            ssum[i] += __shfl_xor(ssum[i], mask, 16);
            sacc[i] += __shfl_xor(sacc[i], mask, 16);
        }
    }

    // lane 0 writes rows tile+0..7, lane 16 writes rows tile+8..15
    if (col == 0) {
        #pragma unroll
        for (int i = 0; i < 8; ++i) {
            int row = tile + half * 8 + i;
            if (row < N) out[row] = sacc[i] / ssum[i];
        }
    }
}

extern "C" void kernel_launch(void* const* d_in, const int* in_sizes, int n_in,
                              void* d_out, int out_size, void* d_ws, size_t ws_size,
                              hipStream_t stream) {
    const float* x       = (const float*)d_in[0];   // (N,3)
    const float* centers = (const float*)d_in[1];   // (C,3)
    const float* beta    = (const float*)d_in[2];   // (C,)
    const float* lin_w   = (const float*)d_in[3];   // (1,C)
    float* out           = (float*)d_out;           // (N,1)

    const int N = in_sizes[0] / 3;
    const int C = in_sizes[2];

    const int rows_per_block = 128;                 // 8 waves * 16 rows
    const int grid = (N + rows_per_block - 1) / rows_per_block;

    rbf_wmma_kernel<<<grid, 256, 0, stream>>>(x, centers, beta, lin_w, out, N, C);
}